// FlattenSELayer_738734375755
// MI455X (gfx1250) — compile-verified
//
#include <hip/hip_runtime.h>
#include <hip/hip_bf16.h>
#include <math.h>

typedef __attribute__((ext_vector_type(2))) float v2f;
typedef __attribute__((ext_vector_type(8))) float v8f;

#define NSEG 16
#define CH   128
#define HID  32

// Workspace layout (floats):
//   [0      , 2048) : segment sums (16 x 128)
//   [2048   , 2064) : segment counts (16)
//   [2064   , 4112) : gate (16 x 128)
#define WS_SUM   0
#define WS_CNT   (NSEG*CH)
#define WS_GATE  (NSEG*CH + NSEG)

// ---------------------------------------------------------------------------
// Zero the accumulator region of the workspace (deterministic per launch).
// ---------------------------------------------------------------------------
__global__ void se_zero_kernel(float* __restrict__ ws) {
    int i = blockIdx.x * blockDim.x + threadIdx.x;
    if (i < NSEG*CH + NSEG) ws[i] = 0.0f;
}

// ---------------------------------------------------------------------------
// Pass 1: segment sums + counts. One wave handles one row per iteration:
// lane = channel group of 4 (float4 / b128 loads). LDS ds_add_f32 accumulate,
// then one global atomic flush per block.
// ---------------------------------------------------------------------------
__global__ void se_segsum_kernel(const float* __restrict__ x,
                                 const int*   __restrict__ idx,
                                 float* __restrict__ ws, int N) {
    __shared__ float ls[NSEG*CH];
    __shared__ float lc[NSEG];
    for (int i = threadIdx.x; i < NSEG*CH; i += blockDim.x) ls[i] = 0.0f;
    if (threadIdx.x < NSEG) lc[threadIdx.x] = 0.0f;
    __syncthreads();

    const int lane  = threadIdx.x & 31;
    const int wave  = (blockIdx.x * blockDim.x + threadIdx.x) >> 5;
    const int nwave = (gridDim.x * blockDim.x) >> 5;

    for (int r = wave; r < N; r += nwave) {
        int s = idx[r];
        const float4 v = ((const float4*)(x + (size_t)r * CH))[lane];
        float* base = ls + s * CH + lane * 4;
        atomicAdd(base + 0, v.x);
        atomicAdd(base + 1, v.y);
        atomicAdd(base + 2, v.z);
        atomicAdd(base + 3, v.w);
        if (lane == 0) atomicAdd(&lc[s], 1.0f);
    }
    __syncthreads();

    for (int i = threadIdx.x; i < NSEG*CH; i += blockDim.x)
        atomicAdd(ws + WS_SUM + i, ls[i]);
    if (threadIdx.x < NSEG)
        atomicAdd(ws + WS_CNT + threadIdx.x, lc[threadIdx.x]);
}

// ---------------------------------------------------------------------------
// Phase 2: pooled = sum/max(cnt,1); gate = sigmoid(relu(pooled*W1^T)*W2^T).
// Single block; wave 0 runs the fp32 WMMA chain (16x16x4_f32 tiles).
//   GEMM1: (16x128)*(128x32) -> 2 N-tiles x 32 K-steps
//   GEMM2: (16x32)*(32x128)  -> 8 N-tiles x  8 K-steps
// A fragments come from LDS; B fragments gathered per-lane from row-major
// weights: B[k][n] = W[n][k].
// ---------------------------------------------------------------------------
__global__ void se_mlp_kernel(const float* __restrict__ W1,   // (32,128)
                              const float* __restrict__ W2,   // (128,32)
                              float* __restrict__ ws) {
    __shared__ float pooled[NSEG*CH];   // 16 x 128
    __shared__ float hbuf[NSEG*HID];    // 16 x 32

    // All threads: compute pooled into LDS.
    for (int i = threadIdx.x; i < NSEG*CH; i += blockDim.x) {
        int seg = i >> 7;
        float c = ws[WS_CNT + seg];
        pooled[i] = ws[WS_SUM + i] / fmaxf(c, 1.0f);
    }
    __syncthreads();

    // Wave-uniform branch: wave 0 fully active -> EXEC all-1s for WMMA.
    if (threadIdx.x < 32) {
        const int lane = threadIdx.x;
        const bool hi  = lane >= 16;
        const int  l16 = lane & 15;
        const int  kph = hi ? 2 : 0;   // K phase for this half-wave

        // ---- GEMM1 + ReLU -> hbuf -------------------------------------
        for (int n0 = 0; n0 < HID; n0 += 16) {
            v8f acc = {};
            for (int k = 0; k < CH; k += 4) {
                int ka = k + kph;
                v2f a, b;
                a.x = pooled[l16 * CH + ka];
                a.y = pooled[l16 * CH + ka + 1];
                b.x = W1[(n0 + l16) * CH + ka];       // W1^T[k][n]
                b.y = W1[(n0 + l16) * CH + ka + 1];
                acc = __builtin_amdgcn_wmma_f32_16x16x4_f32(
                        false, a, false, b, (short)0, acc, false, false);
            }
            // D layout: VGPR v holds row M = v + (hi?8:0), col N = n0+l16
            for (int v = 0; v < 8; ++v) {
                int m = v + (hi ? 8 : 0);
                hbuf[m * HID + n0 + l16] = fmaxf(acc[v], 0.0f);
            }
        }
        // LDS ops from the same wave are in-order; only this wave touches hbuf.

        // ---- GEMM2 + sigmoid -> gate ----------------------------------
        for (int n0 = 0; n0 < CH; n0 += 16) {
            v8f acc = {};
            for (int k = 0; k < HID; k += 4) {
                int ka = k + kph;
                v2f a, b;
                a.x = hbuf[l16 * HID + ka];
                a.y = hbuf[l16 * HID + ka + 1];
                b.x = W2[(n0 + l16) * HID + ka];      // W2^T[k][n]
                b.y = W2[(n0 + l16) * HID + ka + 1];
                acc = __builtin_amdgcn_wmma_f32_16x16x4_f32(
                        false, a, false, b, (short)0, acc, false, false);
            }
            for (int v = 0; v < 8; ++v) {
                int m = v + (hi ? 8 : 0);
                float z = acc[v];
                ws[WS_GATE + m * CH + n0 + l16] = 1.0f / (1.0f + __expf(-z));
            }
        }
    }
}

// ---------------------------------------------------------------------------
// Pass 2: out = x * gate[idx]. Gate cached in LDS; wave-per-row float4 I/O.
// ---------------------------------------------------------------------------
__global__ void se_modulate_kernel(const float* __restrict__ x,
                                   const int*   __restrict__ idx,
                                   const float* __restrict__ gate,
                                   float* __restrict__ out, int N) {
    __shared__ float g[NSEG*CH];
    for (int i = threadIdx.x; i < NSEG*CH; i += blockDim.x) g[i] = gate[i];
    __syncthreads();

    const int lane  = threadIdx.x & 31;
    const int wave  = (blockIdx.x * blockDim.x + threadIdx.x) >> 5;
    const int nwave = (gridDim.x * blockDim.x) >> 5;

    for (int r = wave; r < N; r += nwave) {
        int s = idx[r];
        const float4 v  = ((const float4*)(x + (size_t)r * CH))[lane];
        const float4 gv = ((const float4*)(g + s * CH))[lane];
        float4 o;
        o.x = v.x * gv.x; o.y = v.y * gv.y;
        o.z = v.z * gv.z; o.w = v.w * gv.w;
        ((float4*)(out + (size_t)r * CH))[lane] = o;
    }
}

// ---------------------------------------------------------------------------
extern "C" void kernel_launch(void* const* d_in, const int* in_sizes, int n_in,
                              void* d_out, int out_size, void* d_ws, size_t ws_size,
                              hipStream_t stream) {
    const float* x   = (const float*)d_in[0];
    const int*   idx = (const int*)  d_in[1];
    const float* W1  = (const float*)d_in[2];
    const float* W2  = (const float*)d_in[3];
    float*       out = (float*)d_out;
    float*       ws  = (float*)d_ws;

    const int N = in_sizes[1];          // number of rows / indices

    se_zero_kernel<<<(NSEG*CH + NSEG + 255) / 256, 256, 0, stream>>>(ws);
    se_segsum_kernel<<<1024, 256, 0, stream>>>(x, idx, ws, N);
    se_mlp_kernel<<<1, 128, 0, stream>>>(W1, W2, ws);
    se_modulate_kernel<<<4096, 256, 0, stream>>>(x, idx, ws + WS_GATE, out, N);
}